// DispersedMemory_90013924590022
// MI455X (gfx1250) — compile-verified
//
#include <hip/hip_runtime.h>
#include <stdint.h>

// Problem constants (from the reference)
#define B_   16
#define T_   4096
#define D_   512
// Tiling
#define TT   128                 // t-rows of output per block
#define DT   128                 // floats of D per block (32 float4 columns)
#define HALO 16                  // max |tap offset|
#define ROWS (TT + 2 * HALO)     // 160 LDS rows -> 160*128*4 = 80 KB (dynamic LDS)

typedef __attribute__((ext_vector_type(4))) float        f4;
typedef __attribute__((ext_vector_type(4))) unsigned int uint4v;
typedef __attribute__((ext_vector_type(8))) int          int8v;
typedef __attribute__((ext_vector_type(4))) int          int4v;

__global__ __launch_bounds__(256) void DispersedMemory_kernel(
    const float* __restrict__ P,   // [B, T, D]
    const float* __restrict__ ML,  // [8, D]
    const float* __restrict__ MR,  // [8, D]
    float* __restrict__ out)       // [B, T, D]
{
  extern __shared__ float tile[];  // ROWS * DT floats = 80 KB, rows contiguous

  const int tid = threadIdx.x;
  const int t0  = blockIdx.x * TT;
  const int d0  = blockIdx.y * DT;
  const int b   = blockIdx.z;

  // ---- halo bookkeeping (uniform per block) ----
  int g_start  = t0 - HALO;   // first global t-row we want in LDS row 0
  int lds_row0 = 0;           // LDS row where the DMA starts writing
  int rows     = ROWS;        // rows the DMA transfers
  if (g_start < 0) { lds_row0 = -g_start; rows = ROWS + g_start; g_start = 0; }

  // Front halo of the very first t-tile: zero rows [0,16) = 2048 floats.
  // Disjoint from the TDM destination region, so it can overlap the DMA.
  if (lds_row0 > 0) {
    f4 z = {0.f, 0.f, 0.f, 0.f};
    ((f4*)tile)[tid]       = z;   // 256 thr * 2 * 16 B = 8 KB = 16 rows
    ((f4*)tile)[tid + 256] = z;
  }

  // ---- Tensor Data Mover: DMA the (rows x DT) tile, stride D, into LDS ----
  // TDM ignores EXEC, so the guard MUST be a scalar branch: readfirstlane makes
  // the wave id an SGPR value -> s_cmp + s_cbranch, waves 1..7 never reach the
  // tensor op. Back-edge zero fill is automatic: tensor_dim1 = T - g_start, so
  // tile rows past the end of P read as zero (TDM OOB-load semantics).
  const int wave_first_tid = __builtin_amdgcn_readfirstlane(tid);
  if (wave_first_tid == 0) {  // wave 0 only (scalar condition)
    const uint64_t gaddr =
        (uint64_t)(uintptr_t)(P + ((size_t)b * T_ + (size_t)g_start) * D_ + d0);
    const uint32_t ldsa = (uint32_t)(uintptr_t)(&tile[lds_row0 * DT]);
    const uint32_t dim1 = (uint32_t)(T_ - g_start);

    uint4v g0;
    g0.x = 1u;                                             // count=1 (valid D#)
    g0.y = ldsa;                                           // lds_addr
    g0.z = (uint32_t)gaddr;                                // global_addr[31:0]
    g0.w = (uint32_t)((gaddr >> 32) & 0x01FFFFFFu)         // global_addr[56:32]
         | (2u << 30);                                     // type = 2 ("image")

    int8v g1;
    g1[0] = (int)(2u << 16);                               // data_size = 4 bytes
    g1[1] = (int)(((uint32_t)DT & 0xFFFFu) << 16);         // tensor_dim0 lo16
    g1[2] = (int)((((uint32_t)DT >> 16) & 0xFFFFu)         // tensor_dim0 hi16
                  | ((dim1 & 0xFFFFu) << 16));             // tensor_dim1 lo16
    g1[3] = (int)(((dim1 >> 16) & 0xFFFFu)                 // tensor_dim1 hi16
                  | ((uint32_t)DT << 16));                 // tile_dim0 = 128
    g1[4] = rows;                                          // tile_dim1 (tile_dim2=0)
    g1[5] = D_;                                            // tensor_dim0_stride = 512
    g1[6] = 0;
    g1[7] = 0;

    int4v gz = {0, 0, 0, 0};                               // 2-D: groups 2/3 unused
#if defined(__clang_major__) && (__clang_major__ >= 23)
    int8v gz8 = {0, 0, 0, 0, 0, 0, 0, 0};
    __builtin_amdgcn_tensor_load_to_lds(g0, g1, gz, gz, gz8, 0);
#else
    __builtin_amdgcn_tensor_load_to_lds(g0, g1, gz, gz, 0);
#endif
    __builtin_amdgcn_s_wait_tensorcnt(0);                  // s_wait_tensorcnt 0
  }
  __syncthreads();                                         // tile (incl. DMA) visible

  // ---- compute: each thread owns one float4 d-column and 16 t-rows ----
  const int dcol = tid & 31;   // 32 float4 columns cover DT=128 floats
  const int tgrp = tid >> 5;   // 8 groups * 16 rows = TT=128 rows
  const int dg   = d0 + dcol * 4;

  // 12 tap-weight float4s in registers.
  // Left band:  offset j=1+3k uses mem_left row (7-k), k=0..5
  // Right band: offset j=1+3k uses mem_right row k,    k=0..5
  f4 wl[6], wr[6];
#pragma unroll
  for (int k = 0; k < 6; ++k) {
    wl[k] = *(const f4*)(ML + (size_t)(7 - k) * D_ + dg);
    wr[k] = *(const f4*)(MR + (size_t)k * D_ + dg);
  }

  const f4* tf = (const f4*)tile;  // [row * (DT/4) + col]
#pragma unroll 4
  for (int r = 0; r < TT / 8; ++r) {                       // 16 rows per group
    const int lrow = HALO + tgrp * (TT / 8) + r;           // 16..143
    f4 acc = tf[lrow * (DT / 4) + dcol];                   // identity tap
#pragma unroll
    for (int k = 0; k < 6; ++k) {
      const int j = 1 + 3 * k;
      const f4 a = tf[(lrow - j) * (DT / 4) + dcol];       // P[t-j]
      const f4 c = tf[(lrow + j) * (DT / 4) + dcol];       // P[t+j]
      acc = acc + wl[k] * a + wr[k] * c;                   // contracts to FMAs
    }
    const int t = t0 + tgrp * (TT / 8) + r;
    // Non-temporal store: output is never re-read; keep P resident in L2.
    __builtin_nontemporal_store(acc, (f4*)(out + ((size_t)b * T_ + t) * D_ + dg));
  }
}

extern "C" void kernel_launch(void* const* d_in, const int* in_sizes, int n_in,
                              void* d_out, int out_size, void* d_ws, size_t ws_size,
                              hipStream_t stream) {
  (void)in_sizes; (void)n_in; (void)out_size; (void)d_ws; (void)ws_size;
  const float* P  = (const float*)d_in[0];
  const float* ML = (const float*)d_in[1];
  const float* MR = (const float*)d_in[2];
  float* out = (float*)d_out;

  dim3 grid(T_ / TT, D_ / DT, B_);   // (32, 4, 16) = 2048 blocks
  dim3 block(256);                    // 8 wave32s
  size_t lds_bytes = (size_t)ROWS * DT * sizeof(float);  // 80 KB dynamic LDS
  hipLaunchKernelGGL(DispersedMemory_kernel, grid, block, lds_bytes, stream,
                     P, ML, MR, out);
}